// MSC_Sequence_27187142983839
// MI455X (gfx1250) — compile-verified
//
#include <hip/hip_runtime.h>

// Problem constants (from the reference)
#define BATCH 4096
#define TLEN  2048
#define DDIM  3
#define SDIM  5
#define HDIM  32
#define LSTR  36   // LDS row stride in dwords (bank-conflict-free for our patterns)

typedef __attribute__((ext_vector_type(2))) float v2f;
typedef __attribute__((ext_vector_type(8))) float v8f;

// D(16x16 f32) = A(16x4 f32) x B(4x16 f32) + C
static __device__ __forceinline__ v8f wmma4(v2f a, v2f b, v8f c) {
  return __builtin_amdgcn_wmma_f32_16x16x4_f32(
      /*neg_a=*/false, a, /*neg_b=*/false, b,
      /*c_mod=*/(short)0, c, /*reuse_a=*/false, /*reuse_b=*/false);
}

static __device__ __forceinline__ float tanh_fast(float x) {
#if __has_builtin(__builtin_amdgcn_tanhf)
  return __builtin_amdgcn_tanhf(x);            // native v_tanh_f32 (TRANS, co-execs with WMMA)
#elif __has_builtin(__builtin_amdgcn_exp2f) && __has_builtin(__builtin_amdgcn_rcpf)
  float cx = fminf(fmaxf(x, -16.0f), 16.0f);
  float e  = __builtin_amdgcn_exp2f(cx * 2.8853900817779268f);  // 2*log2(e)
  return (e - 1.0f) * __builtin_amdgcn_rcpf(e + 1.0f);
#else
  return tanhf(x);
#endif
}

__global__ __launch_bounds__(32)
void msc_scan_kernel(const float* __restrict__ delta,  // [B,T,3]
                     const float* __restrict__ s0,     // [B,5]
                     const float* __restrict__ W1,     // [8,32]
                     const float* __restrict__ b1,     // [32]
                     const float* __restrict__ W2,     // [32,32]
                     const float* __restrict__ b2,     // [32]
                     const float* __restrict__ W3,     // [32,5]
                     const float* __restrict__ b3,     // [5]
                     float* __restrict__ out)          // [B,T,5]
{
  __shared__ float lds[16 * LSTR];                     // one wave per block

  const int lane = threadIdx.x & 31;
  const int n16  = lane & 15;   // column index within a 16-wide tile / row index for A rows
  const int g    = lane >> 4;   // lane half (selects K pair in A/B, M half in C/D)
  const int m0   = blockIdx.x * 16;

  // ---- Preload weights into WMMA B-layout registers (done once) ----
  // B chunk (4xK,16xN): lane n16 = column, v0/v1 = K = 2*g + {0,1} within chunk.
  v2f w1b[2][2];
#pragma unroll
  for (int kc = 0; kc < 2; ++kc)
#pragma unroll
    for (int nt = 0; nt < 2; ++nt) {
      const int k = 4 * kc + 2 * g;
      const int n = 16 * nt + n16;
      w1b[kc][nt].x = W1[k * HDIM + n];
      w1b[kc][nt].y = W1[(k + 1) * HDIM + n];
    }
  v2f w2b[8][2];
#pragma unroll
  for (int kc = 0; kc < 8; ++kc)
#pragma unroll
    for (int nt = 0; nt < 2; ++nt) {
      const int k = 4 * kc + 2 * g;
      const int n = 16 * nt + n16;
      w2b[kc][nt].x = W2[k * HDIM + n];
      w2b[kc][nt].y = W2[(k + 1) * HDIM + n];
    }
  v2f w3b[8];
#pragma unroll
  for (int kc = 0; kc < 8; ++kc) {
    const int k = 4 * kc + 2 * g;
    w3b[kc].x = (n16 < SDIM) ? W3[k * SDIM + n16]       : 0.0f;
    w3b[kc].y = (n16 < SDIM) ? W3[(k + 1) * SDIM + n16] : 0.0f;
  }
  // Bias tiles in C/D layout (all 8 row-VGPRs hold the same per-column value).
  v8f b1t[2], b2t[2];
#pragma unroll
  for (int nt = 0; nt < 2; ++nt) {
    const float s1 = b1[16 * nt + n16];
    const float s2 = b2[16 * nt + n16];
#pragma unroll
    for (int i = 0; i < 8; ++i) { b1t[nt][i] = s1; b2t[nt][i] = s2; }
  }
  const float b3s = (n16 < SDIM) ? b3[n16] : 0.0f;

  // ---- Initial state in C/D layout: vgpr i holds row m = i + 8*g, col n16 ----
  v8f sd;
#pragma unroll
  for (int i = 0; i < 8; ++i) {
    const int m = i + 8 * g;
    sd[i] = (n16 < SDIM) ? s0[(size_t)(m0 + m) * SDIM + n16] : 0.0f;
  }
  // Seed LDS state image (cols 0..15; only 0..4 meaningful, rest zero)
#pragma unroll
  for (int i = 0; i < 8; ++i) lds[(i + 8 * g) * LSTR + n16] = sd[i];

  // Per-lane global pointers / offsets
  const float* dptr = delta + (size_t)(m0 + n16) * TLEN * DDIM;  // valid for lane<16
  float* outp = out + (size_t)m0 * TLEN * SDIM;
  unsigned voff[8];
#pragma unroll
  for (int i = 0; i < 8; ++i)
    voff[i] = (unsigned)((i + 8 * g) * TLEN * SDIM + n16);

  // Prime first delta (lanes 0..15 each own one batch row)
  float d0 = 0.f, d1 = 0.f, d2 = 0.f;
  if (lane < 16) { d0 = dptr[0]; d1 = dptr[1]; d2 = dptr[2]; }

  for (int t = 0; t < TLEN; ++t) {
    // delta -> LDS cols 5..7 (state cols 0..4 already present from prev step)
    if (lane < 16) {
      lds[n16 * LSTR + 5] = d0;
      lds[n16 * LSTR + 6] = d1;
      lds[n16 * LSTR + 7] = d2;
    }
    // x = [state, delta] in A layout: chunk0 = cols {2g,2g+1}, chunk1 = cols {4+2g,5+2g}
    const v2f x0 = *(const v2f*)&lds[n16 * LSTR + 2 * g];
    const v2f x1 = *(const v2f*)&lds[n16 * LSTR + 4 + 2 * g];

    // Prefetch next delta while the matmul chain runs
    if (lane < 16 && t + 1 < TLEN) {
      const float* p = dptr + (size_t)(t + 1) * DDIM;
      d0 = p[0]; d1 = p[1]; d2 = p[2];
    }

    // ---- layer 1: h1 = tanh(x @ W1 + b1), bias via C seed ----
    v8f a1[2];
#pragma unroll
    for (int nt = 0; nt < 2; ++nt) {
      a1[nt] = wmma4(x0, w1b[0][nt], b1t[nt]);
      a1[nt] = wmma4(x1, w1b[1][nt], a1[nt]);
    }
    // tanh + D->LDS (transpose staging)
#pragma unroll
    for (int nt = 0; nt < 2; ++nt)
#pragma unroll
      for (int i = 0; i < 8; ++i)
        lds[(i + 8 * g) * LSTR + 16 * nt + n16] = tanh_fast(a1[nt][i]);

    // h1 in A layout: 8 chunks, each one ds_load_b64
    v2f h1a[8];
#pragma unroll
    for (int c = 0; c < 8; ++c)
      h1a[c] = *(const v2f*)&lds[n16 * LSTR + 4 * c + 2 * g];

    // ---- layer 2: h2 = tanh(h1 @ W2 + b2) ----
    v8f a2[2];
#pragma unroll
    for (int nt = 0; nt < 2; ++nt) {
      a2[nt] = wmma4(h1a[0], w2b[0][nt], b2t[nt]);
#pragma unroll
      for (int c = 1; c < 8; ++c)
        a2[nt] = wmma4(h1a[c], w2b[c][nt], a2[nt]);
    }
#pragma unroll
    for (int nt = 0; nt < 2; ++nt)
#pragma unroll
      for (int i = 0; i < 8; ++i)
        lds[(i + 8 * g) * LSTR + 16 * nt + n16] = tanh_fast(a2[nt][i]);

    v2f h2a[8];
#pragma unroll
    for (int c = 0; c < 8; ++c)
      h2a[c] = *(const v2f*)&lds[n16 * LSTR + 4 * c + 2 * g];

    // ---- layer 3: state = state + h2 @ W3 + b3 (residual via C seed) ----
    v8f a3 = wmma4(h2a[0], w3b[0], sd);
#pragma unroll
    for (int c = 1; c < 8; ++c)
      a3 = wmma4(h2a[c], w3b[c], a3);
#pragma unroll
    for (int i = 0; i < 8; ++i) sd[i] = a3[i] + b3s;

    // New state -> LDS (feeds next step's A build); cols 5..15 get overwritten/ignored
#pragma unroll
    for (int i = 0; i < 8; ++i) lds[(i + 8 * g) * LSTR + n16] = sd[i];

    // Emit ys[:, t, :] straight from the D-layout accumulator
    if (n16 < SDIM) {
      const unsigned toff = (unsigned)(t * SDIM);
#pragma unroll
      for (int i = 0; i < 8; ++i) outp[voff[i] + toff] = sd[i];
    }
  }
}

extern "C" void kernel_launch(void* const* d_in, const int* in_sizes, int n_in,
                              void* d_out, int out_size, void* d_ws, size_t ws_size,
                              hipStream_t stream) {
  (void)in_sizes; (void)n_in; (void)out_size; (void)d_ws; (void)ws_size;
  const float* delta = (const float*)d_in[0];
  const float* s0    = (const float*)d_in[1];
  const float* W1    = (const float*)d_in[2];
  const float* b1    = (const float*)d_in[3];
  const float* W2    = (const float*)d_in[4];
  const float* b2    = (const float*)d_in[5];
  const float* W3    = (const float*)d_in[6];
  const float* b3    = (const float*)d_in[7];
  float* out = (float*)d_out;

  dim3 grid(BATCH / 16);   // 256 single-wave workgroups -> spread across SIMDs
  dim3 block(32);          // one wave32 per tile of 16 batch rows
  hipLaunchKernelGGL(msc_scan_kernel, grid, block, 0, stream,
                     delta, s0, W1, b1, W2, b2, W3, b3, out);
}